// NormLoss_18769007083965
// MI455X (gfx1250) — compile-verified
//
#include <hip/hip_runtime.h>
#include <hip/hip_bf16.h>
#include <stdint.h>

// Problem constants (from reference setup_inputs).
#define NB 8          // batch
#define NC 20         // classes
#define NP 200        // prototypes
#define HW 65536      // pixels per image (256*256)
#define PIX_PER_BLOCK 2048
#define CHUNKS (HW / PIX_PER_BLOCK)   // 32 blocks per image
#define WS_STRIDE 64                  // floats per block partial slot

typedef __attribute__((ext_vector_type(2))) float v2f;
typedef __attribute__((ext_vector_type(8))) float v8f;

// ---------------------------------------------------------------------------
// Kernel 1: per-block segment reduction (HBM-bound gather; CDNA5 async path).
//   - async-copy this block's 2048 labels into LDS (ASYNCcnt path)
//   - derive per-class prototype block [start,count] from the identity input
//   - per pixel: gather |activation| over its class's prototype block
//   - bin into LDS ds_add_f32 atomics; write 20 sums + 20 counts to ws
// ---------------------------------------------------------------------------
__global__ __launch_bounds__(256) void NormLoss_accum(
    const float* __restrict__ A,        // [B, HW, P]
    const int*   __restrict__ labels,   // [B, HW]
    const float* __restrict__ ident,    // [P, C]
    float*       __restrict__ ws)       // [gridDim.x, WS_STRIDE]
{
  __shared__ float s_sum[32];
  __shared__ float s_cnt[32];
  __shared__ int   s_start[32];
  __shared__ int   s_np[32];
  __shared__ int   s_lab[PIX_PER_BLOCK];

  const int t  = threadIdx.x;
  const int bi = blockIdx.x;
  const int b  = bi / CHUNKS;
  const int chunk = bi % CHUNKS;
  const long pixbase = (long)b * HW + (long)chunk * PIX_PER_BLOCK;

  if (t < 32) { s_sum[t] = 0.0f; s_cnt[t] = 0.0f; }

  // Async copy labels for this block into LDS.  Each thread issues loads for
  // the slots it will itself consume, so s_wait_asynccnt alone orders its own
  // LDS reads; the __syncthreads below covers the shared tables anyway.
  #pragma unroll
  for (int i = 0; i < PIX_PER_BLOCK / 256; ++i) {
    const int slot = i * 256 + t;
    uint32_t ldsa = (uint32_t)(uintptr_t)(&s_lab[slot]);   // low 32 bits = LDS offset
    const int* gp = labels + pixbase + slot;
    asm volatile("global_load_async_to_lds_b32 %0, %1, off"
                 :: "v"(ldsa), "v"(gp) : "memory");
  }

  // Per-class prototype block from the one-hot identity (contiguous blocks).
  if (t < NC) {
    int st = -1, np = 0;
    for (int p = 0; p < NP; ++p) {
      if (ident[p * NC + t] != 0.0f) { if (st < 0) st = p; ++np; }
    }
    s_start[t] = st; s_np[t] = np;
  }

  asm volatile("s_wait_asynccnt 0x0" ::: "memory");
  __syncthreads();

  #pragma unroll
  for (int i = 0; i < PIX_PER_BLOCK / 256; ++i) {
    const int slot = i * 256 + t;
    const int lab = s_lab[slot] - 1;
    if (lab >= 0 && lab < NC) {
      const int st = s_start[lab];
      const int np = s_np[lab];
      float acc = 0.0f;
      if (st >= 0) {
        const long off = (pixbase + slot) * (long)NP + st;
        if (((st | np | (int)(off & 1)) & 1) == 0) {
          // 8-byte aligned fast path (holds for this input: st=10c, stride 200)
          const float2* ap2 = (const float2*)(A + off);
          for (int j = 0; j < (np >> 1); ++j) {
            float2 v = ap2[j];
            acc += fabsf(v.x) + fabsf(v.y);
          }
        } else {
          const float* ap = A + off;
          for (int j = 0; j < np; ++j) acc += fabsf(ap[j]);
        }
      }
      atomicAdd(&s_sum[lab], acc);
      atomicAdd(&s_cnt[lab], 1.0f);
    }
  }
  __syncthreads();

  if (t < NC) {
    ws[(long)bi * WS_STRIDE + t]      = s_sum[t];
    ws[(long)bi * WS_STRIDE + 32 + t] = s_cnt[t];
  }
}

// ---------------------------------------------------------------------------
// Kernel 2: finalize, single wave (32 threads, EXEC all-ones as WMMA needs).
// The 320-cell x 32-chunk partial reduction is done on the matrix unit:
// D = A x ones + C with V_WMMA_F32_16X16X4_F32 (fp32 FMA, RNE -> same
// numerics as scalar).  A per-lane layout (ISA 7.12.2, 32-bit A 16x4):
//   lanes 0-15 : row M=lane,    VGPR0=K0,   VGPR1=K1
//   lanes 16-31: row M=lane-16, VGPR0=K2,   VGPR1=K3
// B = all-ones so every column of D is the row-sum; extract column N=0 from
// lane 0 (rows 0-7) and lane 16 (rows 8-15) per the C/D layout.
// ---------------------------------------------------------------------------
__global__ __launch_bounds__(32) void NormLoss_finalize(
    const float* __restrict__ ws,
    const float* __restrict__ ident,
    float* __restrict__ out)
{
  __shared__ float s_red[2 * NB * NC];   // [0,160): sums, [160,320): counts
  __shared__ float s_pc[NC];

  const int t = threadIdx.x;       // 0..31, one wave
  const int m = t & 15;            // row within 16-row group
  const int khalf = t >> 4;        // 0: K=0,1   1: K=2,3

  if (t < NC) {
    float pc = 0.0f;
    for (int p = 0; p < NP; ++p) pc += ident[p * NC + t];
    s_pc[t] = pc;
  }

  const v2f ones = {1.0f, 1.0f};
  for (int g = 0; g < (2 * NB * NC) / 16; ++g) {   // 20 groups of 16 cells
    const int idx  = g * 16 + m;                   // 0..319
    const int cell = (idx < NB * NC) ? idx : (idx - NB * NC);
    const int b = cell / NC;
    const int c = cell % NC;
    const int add = (idx < NB * NC) ? 0 : 32;      // sums vs counts slot

    v8f d = {};
    #pragma unroll
    for (int k0 = 0; k0 < CHUNKS; k0 += 4) {
      const int kk = k0 + khalf * 2;
      const long base = ((long)b * CHUNKS + kk) * WS_STRIDE + c + add;
      v2f a = { ws[base], ws[base + WS_STRIDE] };  // K=kk, K=kk+1 for row m
      d = __builtin_amdgcn_wmma_f32_16x16x4_f32(
          /*neg_a=*/false, a, /*neg_b=*/false, ones,
          /*c_mod=*/(short)0, d, /*reuse_a=*/false, /*reuse_b=*/false);
    }
    if (m == 0) {                                  // lane 0 & lane 16, column N=0
      const int mbase = khalf * 8;
      #pragma unroll
      for (int r = 0; r < 8; ++r) s_red[g * 16 + mbase + r] = d[r];
    }
  }
  __syncthreads();

  if (t == 0) {
    float num = 0.0f, den = 0.0f;
    for (int cell = 0; cell < NB * NC; ++cell) {
      const float S   = s_red[cell];
      const float cnt = s_red[NB * NC + cell];
      const float pc  = s_pc[cell % NC];
      const float cn  = S / fmaxf(cnt, 1.0f) / fmaxf(pc, 1.0f);
      const bool valid = (cnt > 0.0f) && (pc > 0.0f);
      num += valid ? cn : 0.0f;
      den += valid ? 1.0f : 0.0f;
    }
    out[0] = num / fmaxf(den, 1.0f);
  }
}

// ---------------------------------------------------------------------------
extern "C" void kernel_launch(void* const* d_in, const int* in_sizes, int n_in,
                              void* d_out, int out_size, void* d_ws, size_t ws_size,
                              hipStream_t stream) {
  const float* A      = (const float*)d_in[0];   // [8, 65536, 200] f32
  const int*   labels = (const int*)d_in[1];     // [8, 256, 256]  int
  const float* ident  = (const float*)d_in[2];   // [200, 20]      f32
  float* out = (float*)d_out;
  float* ws  = (float*)d_ws;                     // 256*64*4 B = 64 KB used

  const int totalPix = in_sizes[1];              // 524288
  const int blocks   = totalPix / PIX_PER_BLOCK; // 256

  NormLoss_accum<<<blocks, 256, 0, stream>>>(A, labels, ident, ws);
  NormLoss_finalize<<<1, 32, 0, stream>>>(ws, ident, out);
}